// GraphMemoryUpdate_36051955483059
// MI455X (gfx1250) — compile-verified
//
#include <hip/hip_runtime.h>
#include <hip/hip_bf16.h>
#include <math.h>

// Problem constants
#define BB 64
#define SS 1024
#define VV 1024
#define DD 512

typedef __attribute__((ext_vector_type(2))) float v2f;
typedef __attribute__((ext_vector_type(4))) float v4f;   // clang vector: OK for nontemporal builtins
typedef __attribute__((ext_vector_type(8))) float v8f;

// ---------------------------------------------------------------------------
// Kernel 1: gate MLP heads via V_WMMA_F32_16X16X4_F32.
//   H = gelu(E @ W1 + b1); g = H @ W2 + b2; gate = sigmoid(g)
// grid = (4, 2): x = 16-row M tile of B=64, y = 0(map)/1(step). 256 thr = 8 waves.
// ---------------------------------------------------------------------------
__device__ __forceinline__ float gelu_tanh(float x) {
    const float k = 0.7978845608028654f;  // sqrt(2/pi)
    float x3 = x * x * x;
    return 0.5f * x * (1.0f + tanhf(k * (x + 0.044715f * x3)));
}

__global__ __launch_bounds__(256)
void gmu_gate_kernel(const float* __restrict__ evidence,
                     const float* __restrict__ mW1, const float* __restrict__ mb1,
                     const float* __restrict__ mW2, const float* __restrict__ mb2,
                     const float* __restrict__ sW1, const float* __restrict__ sb1,
                     const float* __restrict__ sW2, const float* __restrict__ sb2,
                     float* __restrict__ gates /* [2*64]: map then step */) {
    const float* W1 = blockIdx.y ? sW1 : mW1;
    const float* b1 = blockIdx.y ? sb1 : mb1;
    const float* W2 = blockIdx.y ? sW2 : mW2;
    const float* b2 = blockIdx.y ? sb2 : mb2;

    __shared__ float gsum[16];
    const int tid  = threadIdx.x;
    const int wave = tid >> 5;
    const int lane = tid & 31;
    const int m0   = blockIdx.x * 16;

    if (tid < 16) gsum[tid] = 0.0f;
    __syncthreads();

    // WMMA per-lane addressing (ISA 7.12.2):
    // A 16x4 f32: lanes 0-15 -> K={0,1}, lanes 16-31 -> K={2,3}, row M = lane&15
    // B 4x16 f32: lanes 0-15 -> K rows {0,1}, lanes 16-31 -> K rows {2,3}, col N = lane&15
    const int mrow  = lane & 15;
    const int khalf = (lane >> 4) * 2;     // 0 or 2
    const float* Arow = evidence + (size_t)(m0 + mrow) * DD;

    float acc[8];
#pragma unroll
    for (int i = 0; i < 8; ++i) acc[i] = 0.0f;

    // Each wave owns 4 of the 32 N-tiles.
    for (int t = 0; t < 4; ++t) {
        const int n0   = (wave * 4 + t) * 16;
        const int ncol = n0 + (lane & 15);

        v8f c = {0.f, 0.f, 0.f, 0.f, 0.f, 0.f, 0.f, 0.f};
        for (int k = 0; k < DD; k += 4) {
            v2f a, b;
            a.x = Arow[k + khalf];
            a.y = Arow[k + khalf + 1];
            b.x = W1[(size_t)(k + khalf) * DD + ncol];
            b.y = W1[(size_t)(k + khalf + 1) * DD + ncol];
            // D = A*B + C, fp32 matrix pipe (emits v_wmma_f32_16x16x4_f32)
            c = __builtin_amdgcn_wmma_f32_16x16x4_f32(
                    /*neg_a=*/false, a, /*neg_b=*/false, b,
                    /*c_mod=*/(short)0, c, /*reuse_a=*/false, /*reuse_b=*/false);
        }

        const float b1n = b1[ncol];
        const float w2n = W2[ncol];
#pragma unroll
        for (int i = 0; i < 8; ++i) {
            // c[i] = H[m0 + i + 8*(lane>=16)][ncol] pre-bias
            acc[i] += gelu_tanh(c[i] + b1n) * w2n;
        }
    }

    // Reduce over the 16 columns held by each lane half (stays inside halves).
#pragma unroll
    for (int mask = 1; mask <= 8; mask <<= 1) {
#pragma unroll
        for (int i = 0; i < 8; ++i) acc[i] += __shfl_xor(acc[i], mask, 32);
    }

    if ((lane & 15) == 0) {
        const int rbase = (lane >> 4) * 8;   // lane 0 -> rows 0..7, lane 16 -> rows 8..15
#pragma unroll
        for (int i = 0; i < 8; ++i) atomicAdd(&gsum[rbase + i], acc[i]);
    }
    __syncthreads();

    if (tid < 16) {
        float g = gsum[tid] + b2[0];
        gates[blockIdx.y * BB + m0 + tid] = 1.0f / (1.0f + expf(-g));
    }
}

// ---------------------------------------------------------------------------
// Kernel 2: streaming copy (identity part of the scatter-add). 512 MB total,
// b128 vector ops with non-temporal hints so we don't churn the 192 MB L2.
// ---------------------------------------------------------------------------
__global__ __launch_bounds__(256)
void gmu_copy_kernel(const v4f* __restrict__ src, v4f* __restrict__ dst,
                     long n4) {
    long i = (long)blockIdx.x * blockDim.x + threadIdx.x;
    if (i < n4) {
        v4f v = __builtin_nontemporal_load(&src[i]);
        __builtin_nontemporal_store(v, &dst[i]);
    }
}

// ---------------------------------------------------------------------------
// Kernel 3: 64 masked scattered adds (one distinct cell per batch element).
// ---------------------------------------------------------------------------
__global__ void gmu_scatter_kernel(const float* __restrict__ gates,
                                   const int* __restrict__ marker_id,
                                   const int* __restrict__ source_idx,
                                   const unsigned char* __restrict__ source_mask,
                                   const int* __restrict__ tsym_idx,
                                   const unsigned char* __restrict__ tsym_mask,
                                   const int* __restrict__ tval_idx,
                                   const unsigned char* __restrict__ tval_mask,
                                   float* __restrict__ out) {
    int b = threadIdx.x;
    if (b >= BB) return;
    const int mk  = marker_id[b];
    const int sidx = source_idx[b];
    const bool smsk = source_mask[b] != 0;

    const bool map_m  = ((mk == 1) || (mk == 2)) && smsk && (tval_mask[b] != 0);
    const bool step_m = (mk == 3) && smsk && (tsym_mask[b] != 0);

    if (map_m) {
        size_t off = (size_t)b * SS * VV + (size_t)sidx * VV + (size_t)tval_idx[b];
        out[off] += gates[b];
    }
    if (step_m) {
        size_t off = (size_t)BB * SS * VV +
                     (size_t)b * SS * SS + (size_t)sidx * SS + (size_t)tsym_idx[b];
        out[off] += gates[BB + b];
    }
}

// ---------------------------------------------------------------------------
// Launch
// ---------------------------------------------------------------------------
extern "C" void kernel_launch(void* const* d_in, const int* in_sizes, int n_in,
                              void* d_out, int out_size, void* d_ws, size_t ws_size,
                              hipStream_t stream) {
    const float* map_memory  = (const float*)d_in[0];
    const float* step_memory = (const float*)d_in[1];
    const float* evidence    = (const float*)d_in[2];
    const int*   marker_id   = (const int*)d_in[3];
    const int*   source_idx  = (const int*)d_in[4];
    const unsigned char* source_mask = (const unsigned char*)d_in[5];
    const int*   tsym_idx    = (const int*)d_in[6];
    const unsigned char* tsym_mask   = (const unsigned char*)d_in[7];
    const int*   tval_idx    = (const int*)d_in[8];
    const unsigned char* tval_mask   = (const unsigned char*)d_in[9];
    const float* map_W1 = (const float*)d_in[10];
    const float* map_b1 = (const float*)d_in[11];
    const float* map_W2 = (const float*)d_in[12];
    const float* map_b2 = (const float*)d_in[13];
    const float* step_W1 = (const float*)d_in[14];
    const float* step_b1 = (const float*)d_in[15];
    const float* step_W2 = (const float*)d_in[16];
    const float* step_b2 = (const float*)d_in[17];

    float* out   = (float*)d_out;
    float* gates = (float*)d_ws;   // 128 floats: [0,64) map_gate, [64,128) step_gate

    // 1) gates (tiny; WMMA f32 path)
    dim3 ggrid(4, 2);
    gmu_gate_kernel<<<ggrid, 256, 0, stream>>>(evidence,
        map_W1, map_b1, map_W2, map_b2,
        step_W1, step_b1, step_W2, step_b2, gates);

    // 2) streaming identity copy: next_map then next_step (64M floats each)
    const long n4 = (long)BB * SS * VV / 4;          // 16,777,216 float4 per tensor
    const int  blocks = (int)((n4 + 255) / 256);
    gmu_copy_kernel<<<blocks, 256, 0, stream>>>((const v4f*)map_memory,
                                                (v4f*)out, n4);
    gmu_copy_kernel<<<blocks, 256, 0, stream>>>((const v4f*)step_memory,
                                                (v4f*)(out + (size_t)BB * SS * VV), n4);

    // 3) 64 masked scatter-adds (after copy in stream order)
    gmu_scatter_kernel<<<1, 64, 0, stream>>>(gates, marker_id, source_idx, source_mask,
                                             tsym_idx, tsym_mask, tval_idx, tval_mask,
                                             out);
}